// CycleQuantize_38817914421257
// MI455X (gfx1250) — compile-verified
//
#include <hip/hip_runtime.h>
#include <hip/hip_bf16.h>
#include <math.h>

// Problem constants (from reference):
//   z:          (B=4, C*D=256, t=8, h=32, w=32) fp32
//   embeddings: (C=4, K=1024, D=64)             fp32
//   N = B*t*h*w = 32768 flattened points per codebook
// Outputs (concatenated in d_out):
//   quantized_st: 4*256*8*32*32      = 8,388,608 floats
//   dist (C,N,K): 4*32768*1024      = 134,217,728 floats
//   perplexity:                       1 float
#define CB        4
#define KCODES    1024
#define DIM       64
#define NPTS      32768          // per codebook
#define THW       8192           // 8*32*32
#define BSTRIDE   2097152        // 256*8192 floats per batch in z
#define QELEMS    8388608        // quantized output elements
#define DIST_PER_C 33554432ull   // N*K

typedef float v2f __attribute__((ext_vector_type(2)));
typedef float v8f __attribute__((ext_vector_type(8)));

// ---------------------------------------------------------------------------
// Kernel 1: e_sq[c][k] = sum_d emb[c][k][d]^2   (4096 values)
// ---------------------------------------------------------------------------
__global__ __launch_bounds__(256) void esq_kernel(const float* __restrict__ emb,
                                                  float* __restrict__ e_sq) {
    int t = blockIdx.x * blockDim.x + threadIdx.x;   // 0..4095
    if (t >= CB * KCODES) return;
    const float* row = emb + (size_t)t * DIM;
    float s = 0.f;
#pragma unroll
    for (int d = 0; d < DIM; ++d) { float v = row[d]; s += v * v; }
    e_sq[t] = s;
}

// ---------------------------------------------------------------------------
// Kernel 2: main distance GEMM + argmin.
// grid = (256, 4 codebooks), block = 256 (8 waves). Each wave handles one
// 16-row n-block and all 64 k-tiles using V_WMMA_F32_16X16X4_F32.
//
// WMMA layouts (wave32):
//  A 16x4 f32:  lanes0-15: V0=A[m=lane][k=0], V1=A[m=lane][k=1]
//               lanes16-31: V0=A[m=lane-16][k=2], V1=k=3
//  B 4x16 f32 (assumed, mirrors A transposed): lanes0-15: V0=B[0][n=lane],
//               V1=B[1][n]; lanes16-31: V0=B[2][n-16], V1=B[3][n-16]
//  C/D 16x16:   lanes0-15: Vr = D[m=r][n=lane]; lanes16-31: Vr = D[m=8+r][n-16]
// ---------------------------------------------------------------------------
__global__ __launch_bounds__(256) void dist_kernel(const float* __restrict__ z,
                                                   const float* __restrict__ emb,
                                                   const float* __restrict__ e_sq,
                                                   float* __restrict__ dist,
                                                   int* __restrict__ idx_out) {
    const int c    = blockIdx.y;
    const int wave = threadIdx.x >> 5;
    const int lane = threadIdx.x & 31;
    const int g    = lane >> 4;          // half-group (0/1)
    const int m    = lane & 15;          // A row / D column within tile
    const int n0   = (blockIdx.x * 8 + wave) * 16;

    const int b   = n0 >> 13;            // n0 / 8192 (block never crosses b)
    const int thw = n0 & (THW - 1);

    // --- Load A tile (16 rows x 64 dims), this lane's slice: d = 4*dc + 2g {,+1}
    const float* zbase = z + (size_t)b * BSTRIDE + (size_t)c * DIM * THW
                           + (size_t)thw + m;
    v2f va[16];
    float zsq = 0.f;
#pragma unroll
    for (int dc = 0; dc < 16; ++dc) {
        int d0 = 4 * dc + 2 * g;
        float a0 = zbase[(size_t)d0 * THW];
        float a1 = zbase[(size_t)(d0 + 1) * THW];
        va[dc] = (v2f){a0, a1};
        zsq += a0 * a0 + a1 * a1;
    }
    // combine the two half-groups: every lane now has ||z_row||^2 for row m
    zsq += __shfl_xor(zsq, 16, 32);
    // per-lane broadcast of the 8 row norms this lane's D registers cover
    float zsq_rows[8];
#pragma unroll
    for (int r = 0; r < 8; ++r)
        zsq_rows[r] = __shfl(zsq, 8 * g + r, 32);

    const float* ebase    = emb + (size_t)c * KCODES * DIM;
    const float* esq_base = e_sq + c * KCODES;
    float* dbase = dist + (size_t)c * DIST_PER_C + (size_t)n0 * KCODES;

    float vmin[8];
    int   vidx[8];
#pragma unroll
    for (int r = 0; r < 8; ++r) { vmin[r] = INFINITY; vidx[r] = 0; }

    for (int kt = 0; kt < KCODES / 16; ++kt) {
        const int k0 = kt * 16;
        v8f acc = {};
        // B slice for this lane: code (k0+m), dims 4*dc + 2g {,+1}
        const float* eb = ebase + (size_t)(k0 + m) * DIM + 2 * g;
#pragma unroll
        for (int dc = 0; dc < 16; ++dc) {
            float2 t = *(const float2*)(eb + 4 * dc);
            v2f vb = (v2f){t.x, t.y};
            acc = __builtin_amdgcn_wmma_f32_16x16x4_f32(
                false, va[dc], false, vb, (short)0, acc, false, false);
        }
        const float ek = esq_base[k0 + m];
        const int   k  = k0 + m;
#pragma unroll
        for (int r = 0; r < 8; ++r) {
            float dval = zsq_rows[r] + ek - 2.0f * acc[r];
            int row = 8 * g + r;
            __builtin_nontemporal_store(dval,
                dbase + (size_t)row * KCODES + (size_t)k);
            if (dval < vmin[r]) { vmin[r] = dval; vidx[r] = k; }
        }
    }

    // argmin butterfly within each 16-lane half-group (rows 8g..8g+7)
#pragma unroll
    for (int off = 8; off >= 1; off >>= 1) {
#pragma unroll
        for (int r = 0; r < 8; ++r) {
            float ov = __shfl_xor(vmin[r], off, 32);
            int   oi = __shfl_xor(vidx[r], off, 32);
            if (ov < vmin[r] || (ov == vmin[r] && oi < vidx[r])) {
                vmin[r] = ov; vidx[r] = oi;
            }
        }
    }
    if (m == 0) {                       // lanes 0 and 16 write 8 rows each
#pragma unroll
        for (int r = 0; r < 8; ++r)
            idx_out[c * NPTS + n0 + 8 * g + r] = vidx[r];
    }
}

// ---------------------------------------------------------------------------
// Kernel 3: quantized_st = z + (emb[c][idx][d] - z)   (forward == gather,
// but keep the reference's exact fp rounding)
// ---------------------------------------------------------------------------
__global__ __launch_bounds__(256) void gather_kernel(const float* __restrict__ z,
                                                     const float* __restrict__ emb,
                                                     const int* __restrict__ idx,
                                                     float* __restrict__ qout) {
    int o = blockIdx.x * blockDim.x + threadIdx.x;
    if (o >= QELEMS) return;
    int b   = o >> 21;                 // / 2097152
    int rem = o & (BSTRIDE - 1);
    int ch  = rem >> 13;               // channel 0..255
    int thw = rem & (THW - 1);
    int c   = ch >> 6;
    int d   = ch & 63;
    int n   = (b << 13) | thw;
    int k   = idx[c * NPTS + n];
    float q  = emb[((size_t)c * KCODES + k) * DIM + d];
    float zv = z[o];
    qout[o] = zv + (q - zv);
}

// ---------------------------------------------------------------------------
// Kernel 4a/4b/4c: histogram + perplexity
// ---------------------------------------------------------------------------
__global__ __launch_bounds__(256) void zero_counts_kernel(int* __restrict__ counts) {
    int t = blockIdx.x * blockDim.x + threadIdx.x;
    if (t < CB * KCODES) counts[t] = 0;
}

__global__ __launch_bounds__(256) void count_kernel(const int* __restrict__ idx,
                                                    int* __restrict__ counts) {
    int t = blockIdx.x * blockDim.x + threadIdx.x;
    if (t >= CB * NPTS) return;
    int c = t >> 15;                   // / NPTS
    atomicAdd(&counts[c * KCODES + idx[t]], 1);
}

__global__ __launch_bounds__(256) void perplexity_kernel(const int* __restrict__ counts,
                                                         float* __restrict__ out) {
    __shared__ float red[CB][256];
    int t = threadIdx.x;
    float s[CB];
#pragma unroll
    for (int c = 0; c < CB; ++c) s[c] = 0.f;
    for (int k = t; k < KCODES; k += 256) {
#pragma unroll
        for (int c = 0; c < CB; ++c) {
            float p = (float)counts[c * KCODES + k] * (1.0f / (float)NPTS);
            s[c] += p * logf(p + 1e-10f);
        }
    }
#pragma unroll
    for (int c = 0; c < CB; ++c) red[c][t] = s[c];
    __syncthreads();
    for (int off = 128; off > 0; off >>= 1) {
        if (t < off) {
#pragma unroll
            for (int c = 0; c < CB; ++c) red[c][t] += red[c][t + off];
        }
        __syncthreads();
    }
    if (t == 0) {
        float perp = 0.f;
#pragma unroll
        for (int c = 0; c < CB; ++c) perp += expf(-red[c][0]);
        out[0] = perp * (1.0f / (float)CB);
    }
}

// ---------------------------------------------------------------------------
extern "C" void kernel_launch(void* const* d_in, const int* in_sizes, int n_in,
                              void* d_out, int out_size, void* d_ws, size_t ws_size,
                              hipStream_t stream) {
    const float* z   = (const float*)d_in[0];
    const float* emb = (const float*)d_in[1];

    float* out_q    = (float*)d_out;                       // 8,388,608 floats
    float* out_dist = out_q + QELEMS;                      // 134,217,728 floats
    float* out_perp = out_dist + (size_t)CB * DIST_PER_C;  // 1 float

    // workspace layout: e_sq (16KB) | idx (512KB) | counts (16KB) ~= 544KB
    float* ws_esq   = (float*)d_ws;
    int*   ws_idx   = (int*)((char*)d_ws + 16384);
    int*   ws_count = (int*)((char*)d_ws + 16384 + 524288);

    esq_kernel<<<(CB * KCODES + 255) / 256, 256, 0, stream>>>(emb, ws_esq);

    dim3 grid(NPTS / 128, CB);       // 8 waves/block * 16 rows/wave = 128 rows
    dist_kernel<<<grid, 256, 0, stream>>>(z, emb, ws_esq, out_dist, ws_idx);

    gather_kernel<<<QELEMS / 256, 256, 0, stream>>>(z, emb, ws_idx, out_q);

    zero_counts_kernel<<<(CB * KCODES + 255) / 256, 256, 0, stream>>>(ws_count);
    count_kernel<<<(CB * NPTS + 255) / 256, 256, 0, stream>>>(ws_idx, ws_count);
    perplexity_kernel<<<1, 256, 0, stream>>>(ws_count, out_perp);
}